// blendv4_29317446762562
// MI455X (gfx1250) — compile-verified
//
#include <hip/hip_runtime.h>
#include <stdint.h>

#define UHLEN 15
#define BLOCK 256
#define CHUNK 8            // == BLOCK/32 waves; one wave stages one timestep row
#define NEARZERO 1e-6f

static_assert(BLOCK / 32 == CHUNK, "one wave per staged timestep row");

// Bytes per staged row: BLOCK columns * 3 floats * 4 bytes = 3072
#define ROWBYTES (BLOCK * 3 * 4)
#define BUFBYTES (CHUNK * ROWBYTES)

// ---- fast transcendental helpers: single v_trans ops, minimal dep chain ----
__device__ __forceinline__ float xexp2(float x) { return __builtin_amdgcn_exp2f(x); }
__device__ __forceinline__ float xlog2(float x) { return __builtin_amdgcn_logf(x); }
__device__ __forceinline__ float frcp (float x) { return __builtin_amdgcn_rcpf(x); }
__device__ __forceinline__ float fexp (float x) { return xexp2(x * 1.442695040888963f); }
__device__ __forceinline__ float fpow (float x, float y) { return xexp2(y * xlog2(x)); }
// x^(-1/4) == sqrt(rsqrt(x)); two hardware trans ops, exact composition
__device__ __forceinline__ float powm025(float x) {
    return __builtin_amdgcn_sqrtf(__builtin_amdgcn_rsqf(x));
}
// tanh(x) for x >= 0: 1 - 2/(exp(2x)+1); saturates to 1.0 on exp overflow
__device__ __forceinline__ float ftanh_pos(float x) {
    return 1.0f - 2.0f * frcp(fexp(2.0f * x) + 1.0f);
}

__global__ __launch_bounds__(BLOCK) void blend_scan_route_kernel(
    const float* __restrict__ xphy,     // (T, G, 3)
    const float* __restrict__ params,   // (G, 32)
    float* __restrict__ out,            // (T, G)
    int T, int G)
{
    // Double-buffered staging of forcing data
    __shared__ float smem[2][CHUNK][BLOCK * 3];   // 2 * 24KB = 48KB

    const int tid = threadIdx.x;
    const int g0  = blockIdx.x * BLOCK;
    const int g   = g0 + tid;
    const int gc  = (g < G) ? g : (G - 1);        // clamp for param loads
    const int wv  = tid >> 5;                     // wave id (wave32)
    const int ln  = tid & 31;                     // lane id
    const uint64_t xbase  = (uint64_t)(uintptr_t)xphy;
    const uint64_t maxA   = xbase + (uint64_t)T * (uint64_t)G * 12u - 16u;
    const int nchunks = (T + CHUNK - 1) / CHUNK;

    // LDS byte offset of smem base (low 32 bits of the flat address are the
    // LDS offset per the aperture mapping); also keeps smem allocated since
    // its address escapes into the asm below.
    const uint32_t ldsb = (uint32_t)(uintptr_t)(&smem[0][0][0]);

    // ---- async global -> LDS stage of one chunk (8 timestep rows) ----------
    auto issue_chunk = [&](int c, int b) {
        int row = c * CHUNK + wv;
        if (row > T - 1) row = T - 1;             // safety clamp (T%CHUNK==0 normally)
        uint64_t gbyte = xbase + ((uint64_t)row * (uint64_t)G + (uint64_t)g0) * 12u;
        uint32_t lbase = ldsb + (uint32_t)(b * BUFBYTES + wv * ROWBYTES);
        #pragma unroll
        for (int i = 0; i < 6; ++i) {             // 32 lanes * 16B * 6 = 3072B
            uint64_t ga = gbyte + (uint32_t)(ln * 16 + i * 512);
            if (ga > maxA) ga = maxA;             // ragged last block / last row
            uint32_t la = lbase + (uint32_t)(ln * 16 + i * 512);
            asm volatile("global_load_async_to_lds_b128 %0, %1, off"
                         :: "v"(la), "v"(ga) : "memory");
        }
    };

    // Kick off chunk 0 immediately; overlap with parameter setup below.
    issue_chunk(0, 0);

    // ---- per-column parameter descale (sigmoid to physical bounds) ---------
    const float* pp = params + (size_t)gc * 32;
    auto DS = [&](int i, float lo, float hi) {
        float s = frcp(1.0f + fexp(-pp[i]));
        return lo + (hi - lo) * s;
    };
    const float inf_pc      = DS(0,  0.0f, 1.0f);
    const float hbv_beta    = DS(1,  0.5f, 3.0f);
    const float vic_bexp    = DS(2,  0.001f, 3.0f);
    const float hmets_alpha = DS(3,  0.3f, 1.0f);
    const float b1_x3       = DS(4,  20.0f, 300.0f);
    const float b1_bfc      = DS(5,  -3.0f, 0.0f);
    const float b1_bfn      = DS(6,  1.0f, 5.0f);
    const float b1_bfmax    = DS(7,  0.1f, 200.0f);
    const float b1_lam      = DS(8,  1.0f, 20.0f);
    const float b1_thresh   = DS(9,  0.0001f, 0.9999f);
    const float max_perc    = DS(10, 1.0f, 50.0f);
    const float perc_sfc    = DS(11, 0.0001f, 0.9999f);
    const float crise_hbv   = DS(12, 0.1f, 50.0f);
    const float b2_bfmax    = DS(13, 0.1f, 100.0f);
    const float b2_x3       = DS(14, 20.0f, 300.0f);
    const float b2_bfc      = DS(15, -3.0f, 0.0f);
    const float b2_bfn      = DS(16, 1.0f, 5.0f);
    const float Tbf         = DS(17, -5.0f, 2.0f);
    const float Kf          = DS(18, 0.0f, 5.0f);
    const float ddf_min     = DS(19, 1.5f, 3.0f);
    const float ddf_plus    = DS(20, 0.0f, 5.0f);
    const float Kcum        = DS(21, 0.01f, 0.2f);
    const float Tbm         = DS(22, -1.0f, 1.0f);
    const float swi         = DS(23, 0.0f, 0.4f);
    const float fcov        = DS(24, 0.0f, 1.0f);
    const float fspr        = DS(25, 0.0f, 1.0f);
    const float smax1       = DS(26, 50.0f, 500.0f);
    const float smax2       = DS(27, 50.0f, 500.0f);
    const float alpha1      = DS(28, 0.3f, 20.0f);
    const float beta1       = DS(29, 0.01f, 5.0f);
    const float alpha2      = DS(30, 0.5f, 13.0f);
    const float beta2       = DS(31, 0.15f, 1.5f);

    const float canfac    = fcov * (1.0f - fspr);
    const float ten_b1    = xexp2(b1_bfc * 3.321928094887362f);   // 10^bfc
    const float ten_b2    = xexp2(b2_bfc * 3.321928094887362f);
    const float inv_s1m   = 1.0f / smax1;
    const float inv_s2m   = 1.0f / smax2;
    const float inv_x31   = 1.0f / b1_x3;
    const float inv_x32   = 1.0f / b2_x3;
    const float inv_1msfc = 1.0f / (1.0f - perc_sfc);

    // ---- gamma unit-hydrograph weights (length 15, normalized) -------------
    float w1[UHLEN], w2[UHLEN];
    auto GUH = [&](float a, float b, float* w) {
        float lw[UHLEN];
        float mx = -1e30f;
        float ib = 1.0f / b;
        #pragma unroll
        for (int k = 0; k < UHLEN; ++k) {
            float tv = (float)(k + 1);
            lw[k] = (a - 1.0f) * (xlog2(tv) * 0.6931471805599453f) - tv * ib;
            mx = fmaxf(mx, lw[k]);
        }
        float s = 0.0f;
        #pragma unroll
        for (int k = 0; k < UHLEN; ++k) { w[k] = fexp(lw[k] - mx); s += w[k]; }
        float inv = 1.0f / s;
        #pragma unroll
        for (int k = 0; k < UHLEN; ++k) w[k] *= inv;
    };
    GUH(alpha1, beta1, w1);
    GUH(alpha2, beta2, w2);

    // ---- scan state + FIR shift registers (all VGPRs) ----------------------
    float snow = NEARZERO, liq = NEARZERO, cum = NEARZERO, s1 = NEARZERO, s2 = NEARZERO;
    float hs[UHLEN], hb[UHLEN];
    #pragma unroll
    for (int k = 0; k < UHLEN; ++k) { hs[k] = 0.0f; hb[k] = 0.0f; }

    // Drain chunk-0 async loads, publish to all waves.
    asm volatile("s_wait_asynccnt 0" ::: "memory");
    __syncthreads();

    int buf = 0;
    for (int c = 0; c < nchunks; ++c) {
        if (c + 1 < nchunks) issue_chunk(c + 1, buf ^ 1);   // overlap next chunk

        #pragma unroll 1
        for (int tt = 0; tt < CHUNK; ++tt) {
            int t = c * CHUNK + tt;
            if (t >= T) break;

            // Explicit DS loads of this column's staged (prcp, temp, pet).
            uint32_t loff = ldsb + (uint32_t)(buf * BUFBYTES + tt * ROWBYTES + tid * 12);
            float pr, tm, pe;
            asm volatile("ds_load_b32 %0, %3\n\t"
                         "ds_load_b32 %1, %3 offset:4\n\t"
                         "ds_load_b32 %2, %3 offset:8\n\t"
                         "s_wait_dscnt 0"
                         : "=v"(pr), "=v"(tm), "=v"(pe)
                         : "v"(loff)
                         : "memory");

            // snow / canopy
            float rain   = (tm >= 0.0f) ? pr : 0.0f;
            float snowf  = (tm <  0.0f) ? pr : 0.0f;
            float canopy = pe * canfac;
            rain  = fmaxf(rain  - canopy, 0.0f);
            snowf = fmaxf(snowf - canopy, 0.0f);
            float refreeze = fminf(Kf * fmaxf(Tbf - tm, 0.0f), liq);
            snow += snowf + refreeze;
            liq  -= refreeze;
            float ddf  = ddf_min + ddf_plus * (1.0f - fexp(-Kcum * cum));
            float melt = fminf(ddf * fmaxf(tm - Tbm, 0.0f), snow);
            cum  = (snow > NEARZERO) ? (cum + melt) : 0.0f;   // snow BEFORE melt removal
            snow -= melt;
            float tmpw = liq + rain + melt;
            float overflow = fmaxf(tmpw - swi * snow, 0.0f);
            liq = (overflow > NEARZERO) ? (swi * snow) : tmpw;

            // infiltration blend
            float sr = fminf(fmaxf(s1 * inv_s1m, 0.0f), 1.0f);
            float tw = ftanh_pos(overflow * inv_s1m);
            float i_gr4j = smax1 * (1.0f - sr * sr) * tw * frcp(1.0f + sr * tw);
            float i_pc   = overflow * inf_pc;
            float i_hbv  = overflow * (1.0f - fpow(sr, hbv_beta));
            float i_vic  = overflow * fpow(1.0f - sr + 1e-8f, vic_bexp);
            float i_hm   = hmets_alpha * (1.0f - sr) * overflow;
            float infil  = fminf((i_pc + i_gr4j + i_hbv + i_vic + i_hm) * 0.2f, overflow);
            float surface = fmaxf(overflow - infil, 0.0f);
            float soilevap = pe * sr;
            s1 = fminf(fmaxf(s1 + infil - soilevap, NEARZERO), smax1);
            sr = s1 * inv_s1m;

            // baseflow 1 blend
            float q  = s1 * inv_x31;
            float q4 = (q * q) * (q * q);
            float bfa = s1 * (1.0f - powm025(1.0f + q4));
            float bfb = ten_b1 * fpow(s1, b1_bfn);
            float bfcv = b1_bfmax * sr;
            float bfd = b1_bfmax * fpow(sr, b1_bfn);
            float bfe = b1_bfmax * (1.0f - fexp(-b1_lam * fmaxf(sr - b1_thresh, 0.0f)));
            float bf1 = fmaxf(fminf((bfa + bfb + bfcv + bfd + bfe) * 0.2f, s1 - NEARZERO), 0.0f);
            s1 -= bf1;
            sr = s1 * inv_s1m;

            // percolation, capillary rise
            float perc = fmaxf(fminf(max_perc *
                           fminf(fmaxf((sr - perc_sfc) * inv_1msfc, 0.0f), 1.0f),
                           s1 - NEARZERO), 0.0f);
            s1 -= perc;
            float crise = fmaxf(fminf(crise_hbv * (1.0f - s1 * inv_s1m), s2 - NEARZERO), 0.0f);
            s1 = fminf(s1 + crise, smax1);
            s2 = s2 + perc - crise;

            // baseflow 2 blend
            float sr2 = fminf(fmaxf(s2 * inv_s2m, 0.0f), 1.0f);
            float p2  = s2 * inv_x32;
            float p24 = (p2 * p2) * (p2 * p2);
            float b2a = s2 * (1.0f - powm025(1.0f + p24));
            float b2b = ten_b2 * fpow(s2, b2_bfn);
            float b2c = b2_bfmax * sr2;
            float b2d = b2_bfmax * fpow(sr2, b2_bfn);
            float bf2 = fmaxf(fminf((b2a + b2b + b2c + b2d) * 0.25f, s2 - NEARZERO), 0.0f);
            s2 -= bf2;

            // fused causal UH routing: out[t] = sum_k w1[k]*srf[t-k] + w2[k]*(bf1+bf2)[t-k]
            #pragma unroll
            for (int k = UHLEN - 1; k > 0; --k) { hs[k] = hs[k - 1]; hb[k] = hb[k - 1]; }
            hs[0] = surface;
            hb[0] = bf1 + bf2;
            float acc = 0.0f;
            #pragma unroll
            for (int k = 0; k < UHLEN; ++k)
                acc = fmaf(w1[k], hs[k], fmaf(w2[k], hb[k], acc));

            if (g < G) out[(size_t)t * G + g] = acc;
        }

        // drain our async loads for the next buffer, then publish across waves
        asm volatile("s_wait_asynccnt 0" ::: "memory");
        __syncthreads();
        buf ^= 1;
    }
}

extern "C" void kernel_launch(void* const* d_in, const int* in_sizes, int n_in,
                              void* d_out, int out_size, void* d_ws, size_t ws_size,
                              hipStream_t stream) {
    const float* xphy   = (const float*)d_in[0];   // (T, G, 3) float32
    const float* params = (const float*)d_in[1];   // (G, 32) float32
    float* out = (float*)d_out;                    // (T, G, 1) float32

    const int G = in_sizes[1] / 32;
    const int T = in_sizes[0] / (3 * G);
    const int nblk = (G + BLOCK - 1) / BLOCK;

    hipLaunchKernelGGL(blend_scan_route_kernel, dim3(nblk), dim3(BLOCK), 0, stream,
                       xphy, params, out, T, G);
}